// PostProcess_137438953933
// MI455X (gfx1250) — compile-verified
//
#include <hip/hip_runtime.h>
#include <hip/hip_bf16.h>
#include <stdint.h>

// ---------------------------------------------------------------------------
// DETR-style postprocess: sigmoid + per-batch top-K over Q*C scores.
// Monotonic-key histogram select (2 passes; pass 2 hits the 192MB L2),
// sigmoid applied only to the K=100 survivors. Pass 1 streams HBM through
// per-wave async global->LDS staging rings (gfx1250 ASYNCcnt engine) with
// no block barriers in the hot loop.
// ---------------------------------------------------------------------------

#define BINS      4096      // 12-bit key histogram
#define CAP       2048      // max candidates kept per batch
#define KTOP      100
#define HTHREADS  256       // 8 wave32 per block
#define NWAVES    (HTHREADS / 32)
#define DEPTH     4         // async pipeline depth per wave (ring slots)

// monotonic unsigned key: key order == float order (ascending)
__device__ __forceinline__ unsigned fkey(float f) {
  unsigned u = __float_as_uint(f);
  return (u & 0x80000000u) ? ~u : (u | 0x80000000u);
}
__device__ __forceinline__ float keyf(unsigned k) {
  unsigned u = (k & 0x80000000u) ? (k & 0x7FFFFFFFu) : ~k;
  return __uint_as_float(u);
}

// gfx1250 async copy: 16B per lane, global -> LDS, tracked by ASYNCcnt
__device__ __forceinline__ void async_ld_f4(unsigned lds_byte_off,
                                            unsigned gl_byte_off,
                                            const float* base) {
  asm volatile("global_load_async_to_lds_b128 %0, %1, %2"
               :: "v"(lds_byte_off), "v"(gl_byte_off), "s"(base)
               : "memory");
}
#define WAIT_ASYNC(n) asm volatile("s_wait_asynccnt " #n ::: "memory")

// ---------------------------------------------------------------------------
__global__ void pp_zero(unsigned* __restrict__ p, int n) {
  int i = blockIdx.x * blockDim.x + threadIdx.x;
  if (i < n) p[i] = 0u;
}

// ---------------------------------------------------------------------------
// Pass 1: per-batch histogram of 12-bit keys. Blocks = B * S, each covers a
// contiguous chunk of one batch. Per-wave async staging rings, 4 deep;
// each wave waits only on its own ASYNCcnt -> no barriers in the stream loop.
__global__ void pp_hist(const float* __restrict__ logits,
                        unsigned* __restrict__ ghist,
                        int perB_f4, int S, int ch) {
  __shared__ unsigned lh[BINS];
  __shared__ float4 stage[NWAVES * DEPTH * 32];   // 16 KB

  const int tid  = threadIdx.x;
  const int wave = tid >> 5;
  const int lane = tid & 31;
  const int b    = blockIdx.x / S;
  const int s    = blockIdx.x % S;

  for (int i = tid; i < BINS; i += HTHREADS) lh[i] = 0u;
  __syncthreads();   // lh zeroed before any atomics

  const int start = s * ch;
  const int end   = (start + ch < perB_f4) ? (start + ch) : perB_f4;
  const int count = (end > start) ? (end - start) : 0;
  const int iters = count / HTHREADS;             // full tiles of 256 float4

  const float4* bsrc = (const float4*)logits + (size_t)b * perB_f4;
  const float*  base = (const float*)(bsrc + start);   // uniform per block

  // per-lane LDS byte offsets of this wave's DEPTH ring slots
  unsigned lds_slot[DEPTH];
  #pragma unroll
  for (int d = 0; d < DEPTH; ++d)
    lds_slot[d] = (unsigned)(uintptr_t)(&stage[(wave * DEPTH + d) * 32 + lane]);

  // prime the ring: tiles 0..DEPTH-1
  #pragma unroll
  for (int d = 0; d < DEPTH; ++d)
    if (d < iters)
      async_ld_f4(lds_slot[d], (unsigned)((d * HTHREADS + tid) * 16), base);

  for (int it = 0; it < iters; ++it) {
    const int slot = it & (DEPTH - 1);
    if (it + DEPTH <= iters) { WAIT_ASYNC(0x3); }   // oldest tile landed
    else                     { WAIT_ASYNC(0x0); }   // tail: drain
    float4 v = stage[(wave * DEPTH + slot) * 32 + lane];
    atomicAdd(&lh[fkey(v.x) >> 20], 1u);
    atomicAdd(&lh[fkey(v.y) >> 20], 1u);
    atomicAdd(&lh[fkey(v.z) >> 20], 1u);
    atomicAdd(&lh[fkey(v.w) >> 20], 1u);
    // refill the slot we just consumed (reads above already retired)
    if (it + DEPTH < iters)
      async_ld_f4(lds_slot[slot],
                  (unsigned)(((it + DEPTH) * HTHREADS + tid) * 16), base);
  }

  // remainder (< 256 float4): plain loads
  for (int i = start + iters * HTHREADS + tid; i < end; i += HTHREADS) {
    float4 v = bsrc[i];
    atomicAdd(&lh[fkey(v.x) >> 20], 1u);
    atomicAdd(&lh[fkey(v.y) >> 20], 1u);
    atomicAdd(&lh[fkey(v.z) >> 20], 1u);
    atomicAdd(&lh[fkey(v.w) >> 20], 1u);
  }
  __syncthreads();

  unsigned* gh = ghist + (size_t)b * BINS;
  for (int i = tid; i < BINS; i += HTHREADS) {
    unsigned c = lh[i];
    if (c) atomicAdd(&gh[i], c);
  }
}

// ---------------------------------------------------------------------------
// Find threshold bin per batch: smallest top-bin set whose count >= K.
__global__ void pp_thresh(const unsigned* __restrict__ ghist,
                          unsigned* __restrict__ thresh, int K) {
  __shared__ unsigned part[256];
  const int b = blockIdx.x;
  const int t = threadIdx.x;
  const unsigned* h = ghist + (size_t)b * BINS;

  unsigned local[16];
  unsigned s = 0;
  #pragma unroll
  for (int j = 0; j < 16; ++j) {          // reversed order: r=0 is top bin
    int r = t * 16 + j;
    local[j] = h[BINS - 1 - r];
    s += local[j];
  }
  part[t] = s;
  __syncthreads();
  for (int off = 1; off < 256; off <<= 1) {   // Hillis-Steele inclusive scan
    unsigned add = (t >= off) ? part[t - off] : 0u;
    __syncthreads();
    part[t] += add;
    __syncthreads();
  }
  unsigned excl = part[t] - s;
  if (excl < (unsigned)K && excl + s >= (unsigned)K) {   // crossing in my range
    unsigned cum = excl;
    #pragma unroll
    for (int j = 0; j < 16; ++j) {
      cum += local[j];
      if (cum >= (unsigned)K) { thresh[b] = (unsigned)(BINS - 1 - (t * 16 + j)); break; }
    }
  }
}

// ---------------------------------------------------------------------------
// Pass 2 (L2-resident): compact candidates with bin >= threshold bin.
__global__ void pp_collect(const float* __restrict__ logits,
                           const unsigned* __restrict__ thresh,
                           unsigned* __restrict__ cnt,
                           unsigned long long* __restrict__ cand,
                           int perB_f4, int S, int ch) {
  const int b = blockIdx.x / S;
  const int s = blockIdx.x % S;
  const unsigned tb = thresh[b];
  const float4* bsrc = (const float4*)logits + (size_t)b * perB_f4;
  const int start = s * ch;
  const int end   = (start + ch < perB_f4) ? (start + ch) : perB_f4;
  unsigned long long* cb = cand + (size_t)b * CAP;

  for (int i = start + threadIdx.x; i < end; i += HTHREADS) {
    float4 v = bsrc[i];
    float e[4] = {v.x, v.y, v.z, v.w};
    #pragma unroll
    for (int c = 0; c < 4; ++c) {
      unsigned key = fkey(e[c]);
      if ((key >> 20) >= tb) {
        unsigned pos = atomicAdd(&cnt[b], 1u);
        if (pos < CAP) {
          unsigned idx = (unsigned)(i * 4 + c);   // batch-relative flat index
          cb[pos] = ((unsigned long long)key << 32) |
                    (unsigned long long)(0xFFFFFFFFu - idx);
        }
      }
    }
  }
}

// ---------------------------------------------------------------------------
// Final: exact rank of each candidate (key desc, index asc), emit top-K.
__global__ void pp_final(const float* __restrict__ segs,
                         const float* __restrict__ sizes,
                         const unsigned long long* __restrict__ cand,
                         const unsigned* __restrict__ cnt,
                         float* __restrict__ out,
                         int B, int Q, int C, int K) {
  __shared__ unsigned long long lc[CAP];
  const int b = blockIdx.x;
  unsigned cn = cnt[b];
  const int n = (int)((cn < (unsigned)CAP) ? cn : (unsigned)CAP);

  for (int i = threadIdx.x; i < n; i += blockDim.x)
    lc[i] = cand[(size_t)b * CAP + i];
  __syncthreads();

  const float sz = sizes[b];
  const int BK = B * K;

  for (int i = threadIdx.x; i < n; i += blockDim.x) {
    const unsigned long long p = lc[i];
    int rank = 0;
    for (int j = 0; j < n; ++j) rank += (lc[j] > p) ? 1 : 0;
    if (rank < K) {
      unsigned key = (unsigned)(p >> 32);
      unsigned idx = 0xFFFFFFFFu - (unsigned)p;
      int q   = (int)(idx / (unsigned)C);
      int cls = (int)(idx % (unsigned)C);
      float f = keyf(key);
      float score = 1.0f / (1.0f + __expf(-f));
      float cc = segs[((size_t)b * Q + q) * 2 + 0];
      float ww = segs[((size_t)b * Q + q) * 2 + 1];
      int o = b * K + rank;
      out[o]                  = score;                   // scores    [B,K]
      out[BK + o]             = (float)cls;              // labels    [B,K]
      out[2 * BK + 2 * o]     = (cc - 0.5f * ww) * sz;   // segments  [B,K,2]
      out[2 * BK + 2 * o + 1] = (cc + 0.5f * ww) * sz;
      out[4 * BK + o]         = (float)q;                // query_ids [B,K]
    }
  }
}

// ---------------------------------------------------------------------------
extern "C" void kernel_launch(void* const* d_in, const int* in_sizes, int n_in,
                              void* d_out, int out_size, void* d_ws, size_t ws_size,
                              hipStream_t stream) {
  const float* logits = (const float*)d_in[0];   // [B,Q,C]
  const float* segs   = (const float*)d_in[1];   // [B,Q,2]
  const float* sizes  = (const float*)d_in[2];   // [B]

  const int B    = in_sizes[2];
  const int Q    = in_sizes[1] / (2 * B);
  const int perB = in_sizes[0] / B;              // Q*C
  const int C    = perB / Q;
  const int K    = (KTOP < perB) ? KTOP : perB;
  const int perB_f4 = perB / 4;
  const int S  = 10;                             // split blocks per batch
  const int ch = (perB_f4 + S - 1) / S;

  // workspace layout: [cand u64 B*CAP][hist u32 B*BINS][cnt u32 B][thresh u32 B]
  unsigned long long* cand = (unsigned long long*)d_ws;
  unsigned* hist = (unsigned*)((char*)d_ws + (size_t)B * CAP * 8);
  unsigned* cnt  = hist + (size_t)B * BINS;
  unsigned* thr  = cnt + B;

  const int nz = B * BINS + B;                   // hist + cnt contiguous
  pp_zero   <<<(nz + 255) / 256, 256, 0, stream>>>(hist, nz);
  pp_hist   <<<B * S, HTHREADS, 0, stream>>>(logits, hist, perB_f4, S, ch);
  pp_thresh <<<B, 256, 0, stream>>>(hist, thr, K);
  pp_collect<<<B * S, HTHREADS, 0, stream>>>(logits, thr, cnt, cand, perB_f4, S, ch);
  pp_final  <<<B, 256, 0, stream>>>(segs, sizes, cand, cnt, (float*)d_out, B, Q, C, K);
}